// Retrace_9844065042706
// MI455X (gfx1250) — compile-verified
//
#include <hip/hip_runtime.h>
#include <cstdint>
#include <cstddef>

// Problem constants (match reference: B=2048, T=512, D=16, f32)
#define T_DIM   512
#define D_DIM   16
#define ROWS    8                    // batch rows per block
#define CH      16                   // timesteps per staged chunk
#define NT      (ROWS * D_DIM)       // 128 threads / block (4 wave32)
#define NSTAGE  3                    // staged tensors: Eq, Tp, Tq (the j+1 stream)
#define CHUNK_B (ROWS * CH * D_DIM * 4)   // 8192 bytes per tensor per chunk
#define NCH     32                   // ceil(511 / 16)
#define GAMMA_F 0.99f

__global__ void retrace_init_ws(double* ws) {
    if (blockIdx.x == 0 && threadIdx.x == 0) ws[0] = 0.0;
}

__global__ void retrace_finalize(const double* __restrict__ ws, float* __restrict__ out) {
    if (blockIdx.x == 0 && threadIdx.x == 0) {
        // N = B * (T-1) * D = 2048 * 511 * 16
        out[0] = (float)(ws[0] * (1.0 / 16744448.0));
    }
}

__global__ __launch_bounds__(NT) void retrace_scan_kernel(
    const float* __restrict__ Q,    // [B,T,D]
    const float* __restrict__ Eq,   // expected_target_Q [B,T,D]
    const float* __restrict__ Tq,   // target_Q [B,T,D]
    const float* __restrict__ Rw,   // rewards [B,T,D]
    const float* __restrict__ Tp,   // target_policy_probs [B,T,D]
    const float* __restrict__ Bp,   // behaviour_policy_probs [B,T]
    double* __restrict__ ws)
{
    // Double-buffered staging for the (j+1)-indexed tensors: 2 * 3 * 8KB = 48KB LDS
    __shared__ __align__(16) float sbuf[2][NSTAGE][CHUNK_B / 4];

    const int tid = threadIdx.x;
    const int b0  = blockIdx.x * ROWS;
    const uint32_t lds0 = (uint32_t)(uintptr_t)(&sbuf[0][0][0]); // LDS byte offset (addr[31:0])

    // Stage one tensor chunk: [row 0..7][t = tstart..tstart+15][d 0..15] -> LDS, 16B per lane
    auto stage = [&](int p, int tensor, const float* gp, int tstart) {
        const uint32_t lbase = lds0 + (uint32_t)((p * NSTAGE + tensor) * CHUNK_B);
#pragma unroll
        for (int i = 0; i < 4; ++i) {
            uint32_t o      = (uint32_t)(tid * 16 + i * (NT * 16)); // [0, 8192) step 16
            uint32_t row    = o >> 10;                              // 1KB per batch row
            uint32_t within = o & 1023u;
            uint64_t ga = (uint64_t)(uintptr_t)((const char*)gp
                          + (size_t)(b0 + row) * (size_t)(T_DIM * D_DIM * 4)
                          + (size_t)tstart * (D_DIM * 4) + within);
            uint32_t la = lbase + o;
            // CDNA5 async copy: global -> LDS, tracked by ASYNCcnt
            asm volatile("global_load_async_to_lds_b128 %0, %1, off"
                         :: "v"(la), "v"(ga) : "memory");
        }
    };

    const int rb = tid >> 4;     // 0..7 : batch row within block
    const int d  = tid & 15;     // 0..15: feature dim
    const int b  = b0 + rb;

    const float* Qrow  = Q  + (size_t)b * (T_DIM * D_DIM) + d;
    const float* Rwrow = Rw + (size_t)b * (T_DIM * D_DIM) + d;
    const float* Bprow = Bp + (size_t)b * T_DIM;

    float carry = Tq[(size_t)b * (T_DIM * D_DIM) + (size_t)(T_DIM - 1) * D_DIM + d];
    float acc   = 0.0f;

    // Chunk k covers j in [jlo, jhi], processed k = 0..31 (j descending).
    // k=0 -> j in [495,510]; ...; k=31 -> j in [0,14].
    // Staged (j+1)-stream chunk starts at jlo+1 -> max t = 511 (in bounds).
    {   // prefetch chunk 0 into buffer 0
        int jlo = 495;
        stage(0, 0, Eq, jlo + 1);
        stage(0, 1, Tp, jlo + 1);
        stage(0, 2, Tq, jlo + 1);
    }

    for (int k = 0; k < NCH; ++k) {
        const int p = k & 1;
        if (k + 1 < NCH) {
            int jlo_n = 495 - 16 * (k + 1); if (jlo_n < 0) jlo_n = 0;
            stage(p ^ 1, 0, Eq, jlo_n + 1);
            stage(p ^ 1, 1, Tp, jlo_n + 1);
            stage(p ^ 1, 2, Tq, jlo_n + 1);
            // 12 async issues just went out for chunk k+1; <=12 outstanding
            // means all of chunk k's copies have landed (in-order completion).
            asm volatile("s_wait_asynccnt 12" ::: "memory");
        } else {
            asm volatile("s_wait_asynccnt 0" ::: "memory");
        }
        __syncthreads();   // make this wave's landed data visible block-wide

        int jhi = 510 - 16 * k;
        int jlo = jhi - 15; if (jlo < 0) jlo = 0;
        const int len = jhi - jlo + 1;

        const float* sE = &sbuf[p][0][0];
        const float* sP = &sbuf[p][1][0];
        const float* sT = &sbuf[p][2][0];
        const int base = rb * (CH * D_DIM) + d;

        for (int lt = len - 1; lt >= 0; --lt) {
            const int j   = jlo + lt;
            const int idx = base + lt * D_DIM;
            float e  = sE[idx];                       // expected_target_Q[b, j+1, d]
            float tp = sP[idx];                       // target_policy_probs[b, j+1, d]
            float tq = sT[idx];                       // target_Q[b, j+1, d]
            float bp = Bprow[j + 1];                  // behaviour[b, j+1]
            float r  = Rwrow[(size_t)j * D_DIM];      // rewards[b, j, d]
            float Qv = Qrow[(size_t)j * D_DIM];       // Q[b, j, d]

            float cw   = __expf(fminf(tp - bp, 0.0f));           // importance weight
            float q    = fmaf(GAMMA_F, fmaf(cw, carry - tq, e), r);
            float diff = Qv - q;
            acc   = fmaf(diff, diff, acc);
            carry = q;
        }
        __syncthreads();   // all reads of sbuf[p] done before it is re-staged
    }

    // Block reduction of squared-error partials (reuse staging LDS as scratch).
    float* red = &sbuf[0][0][0];
    red[tid] = acc;
    __syncthreads();
#pragma unroll
    for (int s = NT / 2; s > 0; s >>= 1) {
        if (tid < s) red[tid] += red[tid + s];
        __syncthreads();
    }
    if (tid == 0) atomicAdd(ws, (double)red[0]);
}

extern "C" void kernel_launch(void* const* d_in, const int* in_sizes, int n_in,
                              void* d_out, int out_size, void* d_ws, size_t ws_size,
                              hipStream_t stream) {
    // setup_inputs order: Q, expected_target_Q, target_Q, rewards,
    //                     target_policy_probs, behaviour_policy_probs
    const float* Q  = (const float*)d_in[0];
    const float* Eq = (const float*)d_in[1];
    const float* Tq = (const float*)d_in[2];
    const float* Rw = (const float*)d_in[3];
    const float* Tp = (const float*)d_in[4];
    const float* Bp = (const float*)d_in[5];
    float*  out = (float*)d_out;
    double* ws  = (double*)d_ws;

    const int B = in_sizes[5] / T_DIM;        // 2048
    const int nblocks = B / ROWS;             // 256

    retrace_init_ws<<<1, 1, 0, stream>>>(ws);
    retrace_scan_kernel<<<nblocks, NT, 0, stream>>>(Q, Eq, Tq, Rw, Tp, Bp, ws);
    retrace_finalize<<<1, 1, 0, stream>>>(ws, out);
}